// TwoBodySphericalHarmonicElectricFieldTensorEmbed_34325378630297
// MI455X (gfx1250) — compile-verified
//
#include <hip/hip_runtime.h>

typedef float v2f __attribute__((ext_vector_type(2)));
typedef float v4f __attribute__((ext_vector_type(4)));
typedef float v8f __attribute__((ext_vector_type(8)));

#define MULT  32
#define NCOMB 18
#define NW    192          // NUM_IRREPS * MULT
#define KDIM  64
#define EPB   64           // edges per block
#define WSTRIDE 200        // padded LDS stride for the weights tile (banks: 200%64=8)
#define OUT_PER_EDGE (MULT * NCOMB)   // 576

#define SQ3  1.7320508075688772f
#define SQ5  2.2360679774997896f
#define SQ15 3.8729833462074170f

__device__ __forceinline__ void sh9(float x, float y, float z, float* s) {
    float x2 = x * x, y2 = y * y, z2 = z * z;
    s[0] = 1.0f;
    s[1] = SQ3 * x;  s[2] = SQ3 * y;  s[3] = SQ3 * z;
    s[4] = SQ15 * x * z;
    s[5] = SQ15 * x * y;
    s[6] = SQ5 * (y2 - 0.5f * (x2 + z2));
    s[7] = SQ15 * y * z;
    s[8] = 0.5f * SQ15 * (z2 - x2);
}

// W_INDEX = [0, 1,1,1, 2,2,2,2,2, 3, 4,4,4, 5,5,5,5,5]
__device__ __forceinline__ int irrep_of(int c) {
    int base = (c >= 9) ? 3 : 0;
    int k    = (c >= 9) ? (c - 9) : c;
    int ir   = (k == 0) ? 0 : ((k < 4) ? 1 : 2);
    return base + ir;
}

__global__ void __launch_bounds__(128)
tensor_embed_kernel(const float* __restrict__ pos,
                    const int*   __restrict__ edge_index,
                    const int*   __restrict__ batch,
                    const float* __restrict__ elec_field,
                    const float* __restrict__ invar,
                    const float* __restrict__ Wlin,
                    float*       __restrict__ out,
                    int E, int n_frames)
{
    __shared__ float s_field[32 * 9];
    __shared__ float s_sh[EPB * NCOMB];
    __shared__ float s_w[EPB * WSTRIDE];

    const int tid    = threadIdx.x;
    const int e_base = blockIdx.x * EPB;

    // ---- Phase 0a: per-frame field spherical harmonics (normalize=False) ----
    if (tid < 32) {
        if (tid < n_frames) {
            float fx = elec_field[tid * 3 + 0];
            float fy = elec_field[tid * 3 + 1];
            float fz = elec_field[tid * 3 + 2];
            float s[9];
            sh9(fx, fy, fz, s);
            #pragma unroll
            for (int c = 0; c < 9; ++c) s_field[tid * 9 + c] = s[c];
        } else {
            #pragma unroll
            for (int c = 0; c < 9; ++c) s_field[tid * 9 + c] = 0.0f;
        }
    }
    __syncthreads();

    // ---- Phase 0b: per-edge SH (normalized edge vector) + frame broadcast ----
    if (tid < EPB) {
        int e  = e_base + tid;
        int ec = (e < E) ? e : (E - 1);
        int snd = edge_index[ec];          // edge_index[0][e]  (sender)
        int rcv = edge_index[E + ec];      // edge_index[1][e]  (receiver)
        float vx = pos[rcv * 3 + 0] - pos[snd * 3 + 0];
        float vy = pos[rcv * 3 + 1] - pos[snd * 3 + 1];
        float vz = pos[rcv * 3 + 2] - pos[snd * 3 + 2];
        float rn = rsqrtf(vx * vx + vy * vy + vz * vz);
        float s[9];
        sh9(vx * rn, vy * rn, vz * rn, s);
        #pragma unroll
        for (int c = 0; c < 9; ++c) s_sh[tid * NCOMB + c] = s[c];
        int b = batch[snd];
        b = (b < 0) ? 0 : ((b > 31) ? 31 : b);
        #pragma unroll
        for (int c = 0; c < 9; ++c) s_sh[tid * NCOMB + 9 + c] = s_field[b * 9 + c];
    }
    __syncthreads();

    // ---- Phase 1: weights = edge_invariants @ W_lin via V_WMMA_F32_16X16X4_F32 ----
    // Each wave owns a 16-edge x 192-col tile; K=64 -> 16 WMMA k-steps per N-tile.
    const int wave = tid >> 5;
    const int lane = tid & 31;
    const int l15  = lane & 15;
    const int hi   = lane >> 4;          // 0: lanes 0-15, 1: lanes 16-31

    int erow = e_base + wave * 16 + l15;
    if (erow >= E) erow = E - 1;

    // A fragment (16x4 f32): VGPR0 = K0 (lanes 0-15) / K2 (lanes 16-31), VGPR1 = K1/K3.
    // edge_invariants is streamed exactly once -> non-temporal loads keep it out of
    // the reuse set (W_lin / pos / field stay regular-temporal in WGP$/L2).
    const float* arow = invar + (size_t)erow * KDIM + 2 * hi;
    v2f a[16];
    #pragma unroll
    for (int kk = 0; kk < 16; ++kk)
        a[kk] = __builtin_nontemporal_load((const v2f*)(arow + 4 * kk));

    // B fragment (4x16 f32): VGPR0 = row K0/K2 across lane halves, VGPR1 = K1/K3.
    const float* bcol = Wlin + (size_t)(2 * hi) * NW + l15;

    for (int t = 0; t < 12; ++t) {
        v8f acc = {0.f, 0.f, 0.f, 0.f, 0.f, 0.f, 0.f, 0.f};
        const float* bp = bcol + 16 * t;
        #pragma unroll
        for (int kk = 0; kk < 16; ++kk) {
            v2f b;
            b.x = bp[(4 * kk + 0) * NW];
            b.y = bp[(4 * kk + 1) * NW];
            acc = __builtin_amdgcn_wmma_f32_16x16x4_f32(
                /*neg_a=*/false, a[kk], /*neg_b=*/false, b,
                /*c_mod=*/(short)0, acc, /*reuse_a=*/false, /*reuse_b=*/false);
        }
        // D layout (16x16 f32): VGPR j -> row j (lanes 0-15) / row j+8 (lanes 16-31).
        #pragma unroll
        for (int j = 0; j < 8; ++j) {
            int row = j + 8 * hi;
            s_w[(wave * 16 + row) * WSTRIDE + 16 * t + l15] = acc[j];
        }
    }
    __syncthreads();

    // ---- Phase 2: out[e][m][c] = sh[e][c] * w[e][m*6 + irrep(c)] ----
    // 737 MB write-once stream: non-temporal b128 stores so we don't rinse L2.
    int count = E - e_base;
    if (count > EPB) count = EPB;
    const int n4 = count * (OUT_PER_EDGE / 4);          // 576 % 4 == 0
    float* obase = out + (size_t)e_base * OUT_PER_EDGE; // block region is contiguous

    for (int i4 = tid; i4 < n4; i4 += blockDim.x) {
        int base = i4 * 4;
        int e_l  = base / OUT_PER_EDGE;                 // constant across the 4 elems
        int r    = base - e_l * OUT_PER_EDGE;
        v4f v;
        #pragma unroll
        for (int u = 0; u < 4; ++u) {
            int rr = r + u;
            int m  = rr / NCOMB;
            int c  = rr - m * NCOMB;
            int ir = irrep_of(c);
            v[u] = s_sh[e_l * NCOMB + c] * s_w[e_l * WSTRIDE + m * 6 + ir];
        }
        __builtin_nontemporal_store(v, (v4f*)(obase + base));
    }
}

extern "C" void kernel_launch(void* const* d_in, const int* in_sizes, int n_in,
                              void* d_out, int out_size, void* d_ws, size_t ws_size,
                              hipStream_t stream) {
    const float* pos        = (const float*)d_in[0];
    const int*   edge_index = (const int*)  d_in[1];
    const int*   batch      = (const int*)  d_in[2];
    const float* elec_field = (const float*)d_in[3];
    const float* invar      = (const float*)d_in[4];
    const float* Wlin       = (const float*)d_in[5];
    float*       out        = (float*)d_out;

    const int E        = in_sizes[1] / 2;
    const int n_frames = in_sizes[3] / 3;
    const int nblocks  = (E + EPB - 1) / EPB;

    tensor_embed_kernel<<<dim3(nblocks), dim3(128), 0, stream>>>(
        pos, edge_index, batch, elec_field, invar, Wlin, out, E, n_frames);
}